// RoIPool_44762149159081
// MI455X (gfx1250) — compile-verified
//
#include <hip/hip_runtime.h>
#include <hip/hip_bf16.h>
#include <float.h>

// RoI max pooling for MI455X (gfx1250).
// features: (N=2, C=256, H=50, W=50) fp32, rois: (K, 5) fp32 -> out (K, C, 7, 7) fp32.
//
// One block per (roi, 32-channel slab). Stage the RoI's feature sub-region for
// the slab into LDS with CDNA5 async global->LDS copies
// (global_load_async_to_lds_b32 / ASYNCcnt / s_wait_asynccnt), then compute the
// 7x7 bin maxima reading the tile through an address_space(3) pointer so the
// reads lower to ds_load_b32 (not flat). Features (5 MB) are L2-resident on a
// 192 MB L2 part, so this is latency-bound; WMMA has no contraction to exploit.

namespace {
constexpr int   CDIM   = 256;
constexpr int   HDIM   = 50;
constexpr int   WDIM   = 50;
constexpr int   PP     = 7;          // POOLED
constexpr float SCALEF = 0.0625f;
constexpr int   CCH    = 32;         // channels per block
constexpr int   TH     = 20;         // LDS tile height (>= max region height)
constexpr int   TW     = 20;         // LDS tile width  (>= max region width)
}

#define AS3 __attribute__((address_space(3)))

__device__ __forceinline__ int clampi(int v, int lo, int hi) {
  return v < lo ? lo : (v > hi ? hi : v);
}

__device__ __forceinline__ void async_copy_b32(unsigned lds_off, unsigned long long gaddr) {
  // CDNA5 async global->LDS copy: VDST = LDS byte-offset VGPR, VADDR = 64-bit global addr.
  asm volatile("global_load_async_to_lds_b32 %0, %1, off"
               :: "v"(lds_off), "v"(gaddr)
               : "memory");
}

__device__ __forceinline__ void wait_async_zero() {
  asm volatile("s_wait_asynccnt 0x0" ::: "memory");
}

__global__ __launch_bounds__(256) void roipool_gfx1250_kernel(
    const float* __restrict__ feat,   // (N, C, H, W)
    const float* __restrict__ rois,   // (K, 5)
    float* __restrict__ out,          // (K, C, 7, 7)
    int K) {
  __shared__ float tile[CCH * TH * TW];   // 51.2 KB

  const int k   = blockIdx.x;
  const int c0  = blockIdx.y * CCH;
  const int tid = threadIdx.x;
  if (k >= K) return;

  // --- RoI parameters (uniform; compiler keeps these scalar) ---
  const float* r = rois + (size_t)k * 5;
  const int b  = (int)r[0];
  // floor(v*SCALE + 0.5): SCALE is 2^-4 so the multiply is exact in fp32;
  // FMA contraction cannot change the result.
  const int x1 = (int)floorf(__fadd_rn(__fmul_rn(r[1], SCALEF), 0.5f));
  const int y1 = (int)floorf(__fadd_rn(__fmul_rn(r[2], SCALEF), 0.5f));
  const int x2 = (int)floorf(__fadd_rn(__fmul_rn(r[3], SCALEF), 0.5f));
  const int y2 = (int)floorf(__fadd_rn(__fmul_rn(r[4], SCALEF), 0.5f));

  const int roi_wi = (x2 - x1 + 1) > 1 ? (x2 - x1 + 1) : 1;
  const int roi_hi = (y2 - y1 + 1) > 1 ? (y2 - y1 + 1) : 1;
  const float bin_w = (float)roi_wi / (float)PP;
  const float bin_h = (float)roi_hi / (float)PP;

  // Region bounds = union of all bins (same formulas as per-bin at p=0 / p=PP-1)
  const int r0  = clampi(y1, 0, HDIM);                                           // hstart(0)
  const int r1  = clampi((int)ceilf(__fmul_rn((float)PP, bin_h)) + y1, 0, HDIM); // hend(6)
  const int cw0 = clampi(x1, 0, WDIM);                                           // wstart(0)
  const int cw1 = clampi((int)ceilf(__fmul_rn((float)PP, bin_w)) + x1, 0, WDIM); // wend(6)

  const int sh = (r1  - r0 ) < TH ? (r1  - r0 ) : TH;  // staged extent
  const int sw = (cw1 - cw0) < TW ? (cw1 - cw0) : TW;  // (global fallback beyond tile)

  const float* fbase = feat + (size_t)b * CDIM * HDIM * WDIM;
  const unsigned ldsbase = (unsigned)(uintptr_t)(&tile[0]);
  // LDS-addrspace view of the tile for ds_load lowering (low 32 bits of a
  // generic LDS address are the LDS byte offset).
  const AS3 float* ltile = (const AS3 float*)(uintptr_t)ldsbase;

  // --- Stage slab region into LDS: wave -> channel, lane -> column, loop rows.
  //     No runtime integer division anywhere in this nest.
  const int wave = tid >> 5;
  const int lane = tid & 31;
  if (lane < sw) {
    for (int cc = wave; cc < CCH; cc += 8) {
      const float* g0 = fbase + ((size_t)(c0 + cc) * HDIM + r0) * WDIM + cw0 + lane;
      unsigned l0 = ldsbase + (unsigned)((cc * TH * TW + lane) * 4);
      for (int rh = 0; rh < sh; ++rh) {
        async_copy_b32(l0 + (unsigned)(rh * (TW * 4)),
                       (unsigned long long)(uintptr_t)(g0 + rh * WDIM));
      }
    }
  }
  wait_async_zero();     // this wave's async copies complete
  __syncthreads();       // all waves' copies visible in LDS

  // --- Compute 7x7 bin maxima per channel from LDS ---
  for (int idx = tid; idx < CCH * (PP * PP); idx += 256) {
    const int cc  = idx / (PP * PP);          // constant divisor -> magic multiply
    const int bin = idx - cc * (PP * PP);
    const int ph  = bin / PP;
    const int pw  = bin - ph * PP;

    const int hs = clampi((int)floorf(__fmul_rn((float)ph,       bin_h)) + y1, 0, HDIM);
    const int he = clampi((int)ceilf (__fmul_rn((float)(ph + 1), bin_h)) + y1, 0, HDIM);
    const int ws = clampi((int)floorf(__fmul_rn((float)pw,       bin_w)) + x1, 0, WDIM);
    const int we = clampi((int)ceilf (__fmul_rn((float)(pw + 1), bin_w)) + x1, 0, WDIM);

    float v = 0.0f;  // empty bins -> 0 (matches reference)
    if (he > hs && we > ws) {
      float m = -FLT_MAX;  // reference seeds with finfo(float32).min
      for (int h = hs; h < he; ++h) {
        const int rh = h - r0;                               // >= 0 by construction
        const AS3 float* lrow = ltile + (cc * TH + rh) * TW;
        const float* grow = fbase + ((size_t)(c0 + cc) * HDIM + h) * WDIM;
        for (int w = ws; w < we; ++w) {
          const int rw = w - cw0;                            // >= 0 by construction
          const float x = (rh < sh && rw < sw) ? lrow[rw] : grow[w];
          m = fmaxf(m, x);
        }
      }
      v = m;
    }
    out[((size_t)k * CDIM + (c0 + cc)) * (PP * PP) + bin] = v;
  }
}

extern "C" void kernel_launch(void* const* d_in, const int* in_sizes, int n_in,
                              void* d_out, int out_size, void* d_ws, size_t ws_size,
                              hipStream_t stream) {
  const float* feat = (const float*)d_in[0];
  const float* rois = (const float*)d_in[1];
  float* out        = (float*)d_out;
  const int K = in_sizes[1] / 5;

  dim3 grid((unsigned)K, (unsigned)(CDIM / CCH));   // (256, 8) blocks x 256 threads
  roipool_gfx1250_kernel<<<grid, 256, 0, stream>>>(feat, rois, out, K);
}